// FilterLayerM1_56599079027318
// MI455X (gfx1250) — compile-verified
//
#include <hip/hip_runtime.h>

// FilterLayer (MoE spectral filter) for MI455X / gfx1250.
//   B=256, S=512, H=256, E=8, F=257
// Pipeline: gate_kernel -> fft_filter_kernel (WMMA matmul-FFT + TDM tile
// load/store) -> ln_kernel (in-place LayerNorm).

#define PI_F 3.14159265358979323846f

typedef float v2f __attribute__((ext_vector_type(2)));
typedef float v8f __attribute__((ext_vector_type(8)));
typedef unsigned int u32x4 __attribute__((ext_vector_type(4)));
typedef int i32x4 __attribute__((ext_vector_type(4)));
typedef int i32x8 __attribute__((ext_vector_type(8)));

static __device__ __forceinline__ v8f wmma4(v2f a, v2f b, v8f c) {
  // D(16x16,f32) = A(16x4,f32) * B(4x16,f32) + C
  return __builtin_amdgcn_wmma_f32_16x16x4_f32(false, a, false, b, (short)0, c,
                                               false, false);
}

// Tensor Data Mover: move a 2-D tile (tile_dim0=16 floats contiguous,
// tile_dim1=512 rows, row stride 256 floats) between global and LDS.
// D# layout per CDNA5 ISA section 8.3/8.4. 2-D tensor => groups 2/3 zero.
// This toolchain (clang-23 / therock-10.0) uses the 6-arg builtin form.
static __device__ __forceinline__ void tdm_tile_2d(bool store, uint32_t lds_off,
                                                   const void* gptr) {
  const uint64_t ga = (uint64_t)(uintptr_t)gptr;
  u32x4 g0;
  g0.x = 1u;             // count=1, is_restore=0, gather off
  g0.y = lds_off;        // LDS byte address of tile
  g0.z = (uint32_t)ga;   // global_addr[31:0]
  g0.w = ((uint32_t)(ga >> 32) & 0x01FFFFFFu) | (2u << 30);  // [56:32] | type=2
  i32x8 g1;
  g1[0] = 0x00020000;    // workgroup_mask=0, data_size=2 (4B), no pad/iterate
  g1[1] = (16 << 16);    // tensor_dim0 = 16 (elements)
  g1[2] = (512 << 16);   // tensor_dim0[31:16]=0 | tensor_dim1 = 512
  g1[3] = (16 << 16);    // tensor_dim1[31:16]=0 | tile_dim0 = 16
  g1[4] = 512;           // tile_dim1 = 512 | tile_dim2 = 0
  g1[5] = 256;           // tensor_dim0_stride = 256 (elements)
  g1[6] = 0;             // stride0[47:32]=0 | tensor_dim1_stride lo = 0
  g1[7] = 0;
  const i32x4 z4 = {0, 0, 0, 0};
  const i32x8 z8 = {0, 0, 0, 0, 0, 0, 0, 0};
  if (store)
    __builtin_amdgcn_tensor_store_from_lds(g0, g1, z4, z4, z8, 0);
  else
    __builtin_amdgcn_tensor_load_to_lds(g0, g1, z4, z4, z8, 0);
}

// ---------------------------------------------------------------------------
// Kernel A: gate[b, e] = softmax_e( (mean_s x * 50/(valid+1e-4)) @ mlp_w + b )
// One block per batch, 256 threads.
// ---------------------------------------------------------------------------
__global__ __launch_bounds__(256) void gate_kernel(
    const float* __restrict__ x, const float* __restrict__ mw,
    const float* __restrict__ mb, float* __restrict__ gate) {
  __shared__ float red[256];
  __shared__ int redi[256];
  __shared__ float logits[8];
  __shared__ float scale_s;

  const int tid = threadIdx.x;
  const int b = blockIdx.x;
  const float* xb = x + (size_t)b * 512 * 256;

  // valid-length: count of s-rows whose H-sum != 0 (thread owns rows tid, tid+256)
  float rs0 = 0.f, rs1 = 0.f;
  for (int h = 0; h < 256; ++h) {
    rs0 += xb[tid * 256 + h];
    rs1 += xb[(tid + 256) * 256 + h];
  }
  redi[tid] = (rs0 != 0.f ? 1 : 0) + (rs1 != 0.f ? 1 : 0);

  // per-channel sum over s (thread owns channel tid; coalesced)
  float acc = 0.f;
  for (int s = 0; s < 512; ++s) acc += xb[s * 256 + tid];
  __syncthreads();

  for (int off = 128; off > 0; off >>= 1) {
    if (tid < off) redi[tid] += redi[tid + off];
    __syncthreads();
  }
  if (tid == 0) scale_s = 50.0f / (((float)redi[0] + 1e-4f) * 512.0f);
  __syncthreads();

  const float m = acc * scale_s;  // gate pre-matmul value for channel tid
  for (int e = 0; e < 8; ++e) {
    red[tid] = m * mw[tid * 8 + e];
    __syncthreads();
    for (int off = 128; off > 0; off >>= 1) {
      if (tid < off) red[tid] += red[tid + off];
      __syncthreads();
    }
    if (tid == 0) logits[e] = red[0] + mb[e];
    __syncthreads();
  }
  if (tid == 0) {
    float mx = logits[0];
    for (int e = 1; e < 8; ++e) mx = fmaxf(mx, logits[e]);
    float ex[8], se = 0.f;
    for (int e = 0; e < 8; ++e) { ex[e] = expf(logits[e] - mx); se += ex[e]; }
    for (int e = 0; e < 8; ++e) gate[b * 8 + e] = ex[e] / se;
  }
}

// ---------------------------------------------------------------------------
// Kernel C: per (b, 16-wide h tile): 512-pt complex FFT along S via WMMA
// matmuls (512 = 16 x 32 four-step), multiply by Hermitian-extended mixed
// filter, inverse FFT, residual add. TDM loads x tile / stores h_pre tile.
//
// Index maps: n = n1*32 + n2,  k = k1 + 16*k2   (n1,k1 in [0,16); n2,k2 in [0,32))
//   fwd:  U[k1,n2] = sum_n1 D16[k1,n1] x[n1*32+n2]   (D16 = exp(-2pi i k n/16))
//         T = tw(k1,n2) * U,  tw = exp(-2pi i k1 n2/512)
//         X[k1+16k2] = sum_n2 D32[k2,n2] T[k1,n2]
//   inv (per k1):  V[n2] = sum_k2 conj(D32)[n2,k2] (X*wext)[k1+16k2]
//         V' = conj(tw) * V
//         y[n1*32+n2] = Re( sum_k1 conj(D16)[n1,k1] V'[k1,n2] ) / 512
// ---------------------------------------------------------------------------
__global__ __launch_bounds__(256) void fft_filter_kernel(
    const float* __restrict__ x, const float* __restrict__ cw,
    const float* __restrict__ gate, float* __restrict__ out) {
  extern __shared__ __align__(16) float sm[];
  float* xs = sm;            // [512][16] input tile / output tile
  float* TrB = sm + 8192;    // [16(k1)][32(n2)][16(h)] real
  float* TiB = sm + 16384;   // imag
  float* Xw = sm + 24576;    // per-wave scratch: 8 waves * (512 re + 512 im)
  float* d16r = sm + 32768;  // [16][16]
  float* d16i = sm + 33024;
  float* d32r = sm + 33280;  // [32][32]
  float* d32i = sm + 34304;
  float* twr = sm + 35328;   // [16][32]
  float* twi = sm + 35840;   // total 36352 floats = 145408 B

  const int tid = threadIdx.x;
  const int lane = tid & 31;
  const int wave = tid >> 5;
  const int hi = (lane >> 4) & 1;  // which half of the wave
  const int hl = lane & 15;        // h (N) column within tile
  const int b = blockIdx.x >> 4;
  const int h0 = (blockIdx.x & 15) << 4;
  const uint32_t xs_off = (uint32_t)(uintptr_t)xs;

  // ---- kick off async DMA of the x tile (overlaps table build) ----
  if (wave == 0)
    tdm_tile_2d(false, xs_off, x + (size_t)(b * 512) * 256 + h0);

  // ---- build DFT / twiddle tables ----
  {
    int k = tid >> 4, n = tid & 15;
    float ang = -2.0f * PI_F * (float)((k * n) & 15) * (1.0f / 16.0f);
    d16r[tid] = cosf(ang);
    d16i[tid] = sinf(ang);
  }
  for (int i = tid; i < 1024; i += 256) {
    int k = i >> 5, n = i & 31;
    float ang = -2.0f * PI_F * (float)((k * n) & 31) * (1.0f / 32.0f);
    d32r[i] = cosf(ang);
    d32i[i] = sinf(ang);
  }
  for (int i = tid; i < 512; i += 256) {
    int k1 = i >> 5, n2 = i & 31;
    float ang = -2.0f * PI_F * (float)((k1 * n2) & 511) * (1.0f / 512.0f);
    twr[i] = cosf(ang);
    twi[i] = sinf(ang);
  }
  float g[8];
#pragma unroll
  for (int e = 0; e < 8; ++e) g[e] = gate[b * 8 + e];

  if (wave == 0) __builtin_amdgcn_s_wait_tensorcnt(0);
  __syncthreads();

  // ---- preload A fragments (constant DFT matrices) ----
  // A layout (16x4): lanes 0-15 -> M=lane, a.x:K=0 a.y:K=1; lanes 16-31 -> K=2,3
  v2f A32r[2][8], A32i[2][8], A16r[4], A16i[4];
#pragma unroll
  for (int m = 0; m < 2; ++m)
#pragma unroll
    for (int ks = 0; ks < 8; ++ks) {
      int row = m * 16 + hl, col = 4 * ks + 2 * hi;
      v2f t; t.x = d32r[row * 32 + col]; t.y = d32r[row * 32 + col + 1];
      A32r[m][ks] = t;
      v2f u; u.x = d32i[row * 32 + col]; u.y = d32i[row * 32 + col + 1];
      A32i[m][ks] = u;
    }
#pragma unroll
  for (int ks = 0; ks < 4; ++ks) {
    int col = 4 * ks + 2 * hi;
    v2f t; t.x = d16r[hl * 16 + col]; t.y = d16r[hl * 16 + col + 1];
    A16r[ks] = t;
    v2f u; u.x = d16i[hl * 16 + col]; u.y = d16i[hl * 16 + col + 1];
    A16i[ks] = u;
  }
  const v8f vzero = {0.f, 0.f, 0.f, 0.f, 0.f, 0.f, 0.f, 0.f};

  // ---- forward stage 1: U = D16 * x (real input), twiddle, store T ----
  for (int ii = 0; ii < 4; ++ii) {
    int n2 = wave * 4 + ii;
    v2f Bx[4];
#pragma unroll
    for (int ks = 0; ks < 4; ++ks) {  // B rows = n1
      int r0 = 4 * ks + 2 * hi;
      v2f t;
      t.x = xs[(r0 * 32 + n2) * 16 + hl];
      t.y = xs[((r0 + 1) * 32 + n2) * 16 + hl];
      Bx[ks] = t;
    }
    v8f Ur = vzero, Ui = vzero;
#pragma unroll
    for (int ks = 0; ks < 4; ++ks) Ur = wmma4(A16r[ks], Bx[ks], Ur);
#pragma unroll
    for (int ks = 0; ks < 4; ++ks) Ui = wmma4(A16i[ks], Bx[ks], Ui);
#pragma unroll
    for (int r = 0; r < 8; ++r) {  // C rows = k1
      int k1 = r + 8 * hi;
      float tr = twr[k1 * 32 + n2], ti = twi[k1 * 32 + n2];
      float ur = Ur[r], ui = Ui[r];
      TrB[(k1 * 32 + n2) * 16 + hl] = ur * tr - ui * ti;
      TiB[(k1 * 32 + n2) * 16 + hl] = ur * ti + ui * tr;
    }
  }
  __syncthreads();

  // ---- per-k1: fwd stage 3, filter multiply, inverse stage A (in place) ----
  float* XwR = Xw + wave * 1024;
  float* XwI = XwR + 512;
  for (int ii = 0; ii < 2; ++ii) {
    int k1 = wave * 2 + ii;
    v2f Br[8], Bi[8];
#pragma unroll
    for (int ks = 0; ks < 8; ++ks) {  // B rows = n2
      int r0 = 4 * ks + 2 * hi;
      v2f t; t.x = TrB[(k1 * 32 + r0) * 16 + hl];
      t.y = TrB[(k1 * 32 + r0 + 1) * 16 + hl];
      Br[ks] = t;
      v2f u; u.x = TiB[(k1 * 32 + r0) * 16 + hl];
      u.y = TiB[(k1 * 32 + r0 + 1) * 16 + hl];
      Bi[ks] = u;
    }
#pragma unroll
    for (int m = 0; m < 2; ++m) {
      v8f RR = vzero, II = vzero, RI = vzero, IR = vzero;
#pragma unroll
      for (int ks = 0; ks < 8; ++ks) {
        RR = wmma4(A32r[m][ks], Br[ks], RR);
        II = wmma4(A32i[m][ks], Bi[ks], II);
        RI = wmma4(A32r[m][ks], Bi[ks], RI);
        IR = wmma4(A32i[m][ks], Br[ks], IR);
      }
#pragma unroll
      for (int r = 0; r < 8; ++r) {  // C rows = k2
        int k2 = m * 16 + r + 8 * hi;
        int k = k1 + 16 * k2;
        float xr = RR[r] - II[r];
        float xi = RI[r] + IR[r];
        int f = (k <= 256) ? k : (512 - k);
        float wr = 0.f, wi = 0.f;
#pragma unroll
        for (int e = 0; e < 8; ++e) {  // mix experts (L2-resident table)
          float2 wv = reinterpret_cast<const float2*>(
              cw)[(size_t)(e * 257 + f) * 256 + h0 + hl];
          wr += g[e] * wv.x;
          wi += g[e] * wv.y;
        }
        if (k > 256) wi = -wi;  // Hermitian extension: conj(w[512-k])
        XwR[k2 * 16 + hl] = xr * wr - xi * wi;
        XwI[k2 * 16 + hl] = xr * wi + xi * wr;
      }
    }
    // inverse stage A: V[n2] = sum_k2 conj(D32)[n2,k2] * P[k2] (D32 symmetric)
    v2f Cr[8], Ci[8];
#pragma unroll
    for (int ks = 0; ks < 8; ++ks) {  // B rows = k2
      int r0 = 4 * ks + 2 * hi;
      v2f t; t.x = XwR[r0 * 16 + hl]; t.y = XwR[(r0 + 1) * 16 + hl];
      Cr[ks] = t;
      v2f u; u.x = XwI[r0 * 16 + hl]; u.y = XwI[(r0 + 1) * 16 + hl];
      Ci[ks] = u;
    }
#pragma unroll
    for (int m = 0; m < 2; ++m) {
      v8f P1 = vzero, P2 = vzero, P3 = vzero, P4 = vzero;
#pragma unroll
      for (int ks = 0; ks < 8; ++ks) {
        P1 = wmma4(A32r[m][ks], Cr[ks], P1);
        P2 = wmma4(A32i[m][ks], Ci[ks], P2);
        P3 = wmma4(A32r[m][ks], Ci[ks], P3);
        P4 = wmma4(A32i[m][ks], Cr[ks], P4);
      }
#pragma unroll
      for (int r = 0; r < 8; ++r) {  // C rows = n2
        int n2 = m * 16 + r + 8 * hi;
        float vr = P1[r] + P2[r];   // Re = d32r*Xr + d32i*Xi
        float vi = P3[r] - P4[r];   // Im = d32r*Xi - d32i*Xr
        float tr = twr[k1 * 32 + n2], ti = twi[k1 * 32 + n2];
        TrB[(k1 * 32 + n2) * 16 + hl] = vr * tr + vi * ti;  // * conj(tw)
        TiB[(k1 * 32 + n2) * 16 + hl] = vi * tr - vr * ti;
      }
    }
  }
  __syncthreads();

  // ---- inverse stage B: y = Re( conj(D16) * V' ) / 512, residual into xs ----
  for (int ii = 0; ii < 4; ++ii) {
    int n2 = wave * 4 + ii;
    v2f Vr[4], Vi[4];
#pragma unroll
    for (int ks = 0; ks < 4; ++ks) {  // B rows = k1
      int r0 = 4 * ks + 2 * hi;
      v2f t; t.x = TrB[(r0 * 32 + n2) * 16 + hl];
      t.y = TrB[((r0 + 1) * 32 + n2) * 16 + hl];
      Vr[ks] = t;
      v2f u; u.x = TiB[(r0 * 32 + n2) * 16 + hl];
      u.y = TiB[((r0 + 1) * 32 + n2) * 16 + hl];
      Vi[ks] = u;
    }
    v8f Q1 = vzero, Q2 = vzero;
#pragma unroll
    for (int ks = 0; ks < 4; ++ks) {
      Q1 = wmma4(A16r[ks], Vr[ks], Q1);  // d16r * V'r
      Q2 = wmma4(A16i[ks], Vi[ks], Q2);  // d16i * V'i  (Re(conj(D)V)=ac+bd)
    }
#pragma unroll
    for (int r = 0; r < 8; ++r) {  // C rows = n1
      int n1 = r + 8 * hi;
      int s = n1 * 32 + n2;
      float y = (Q1[r] + Q2[r]) * (1.0f / 512.0f);
      xs[s * 16 + hl] = y + xs[s * 16 + hl];  // h_pre tile (lane-private elem)
    }
  }
  __syncthreads();

  // ---- async DMA of h_pre tile back to global; S_ENDPGM waits for idle ----
  if (wave == 0)
    tdm_tile_2d(true, xs_off, out + (size_t)(b * 512) * 256 + h0);
}

// ---------------------------------------------------------------------------
// Kernel D: in-place LayerNorm over H=256. One wave32 per (b,s) row.
// ---------------------------------------------------------------------------
__global__ __launch_bounds__(256) void ln_kernel(float* __restrict__ out,
                                                 const float* __restrict__ gm,
                                                 const float* __restrict__ bt) {
  const int lane = threadIdx.x & 31;
  const int wave = threadIdx.x >> 5;
  const size_t row = (size_t)blockIdx.x * 8 + wave;
  float* p = out + row * 256;

  float v[8];
#pragma unroll
  for (int j = 0; j < 8; ++j) v[j] = p[j * 32 + lane];

  float s = 0.f;
#pragma unroll
  for (int j = 0; j < 8; ++j) s += v[j];
  for (int m = 16; m > 0; m >>= 1) s += __shfl_xor(s, m, 32);
  const float mean = s * (1.0f / 256.0f);

  float q = 0.f;
#pragma unroll
  for (int j = 0; j < 8; ++j) {
    float d = v[j] - mean;
    q += d * d;
  }
  for (int m = 16; m > 0; m >>= 1) q += __shfl_xor(q, m, 32);
  const float inv = 1.0f / sqrtf(q * (1.0f / 256.0f) + 1e-12f);

#pragma unroll
  for (int j = 0; j < 8; ++j) {
    int h = j * 32 + lane;
    p[h] = (v[j] - mean) * inv * gm[h] + bt[h];
  }
}

// ---------------------------------------------------------------------------
extern "C" void kernel_launch(void* const* d_in, const int* in_sizes, int n_in,
                              void* d_out, int out_size, void* d_ws,
                              size_t ws_size, hipStream_t stream) {
  const float* x = (const float*)d_in[0];
  const float* cw = (const float*)d_in[1];   // [8][257][256][2]
  const float* mw = (const float*)d_in[2];   // [256][8]
  const float* mb = (const float*)d_in[3];   // [8]
  const float* gm = (const float*)d_in[4];   // [256]
  const float* bt = (const float*)d_in[5];   // [256]
  float* out = (float*)d_out;                // [256][512][256]
  float* gate = (float*)d_ws;                // [256][8]

  gate_kernel<<<256, 256, 0, stream>>>(x, mw, mb, gate);
  fft_filter_kernel<<<4096, 256, 36352 * sizeof(float), stream>>>(x, cw, gate,
                                                                  out);
  ln_kernel<<<16384, 256, 0, stream>>>(out, gm, bt);
}